// MFE_Loss_54606214202053
// MI455X (gfx1250) — compile-verified
//
#include <hip/hip_runtime.h>

typedef __attribute__((ext_vector_type(2))) float v2f;
typedef __attribute__((ext_vector_type(8))) float v8f;

#define NUM_CLASSES 5
#define GRID 1024
#define BLOCK 256

// Stage 1: streaming log-softmax CE + per-class segment partials.
// Each thread owns groups of 4 consecutive rows: 4 rows x 5 f32 = 80 B
// = five 16B-aligned float4 loads (global_load_b128) + one int4 target load.
// Each block writes one 16-float row to `partials`:
//   [sum0..sum4, cnt0..cnt4, 0,0,0,0,0,0]
__global__ void __launch_bounds__(BLOCK) mfe_partials_kernel(
    const float* __restrict__ logits, const int* __restrict__ targets,
    float* __restrict__ partials, int N)
{
    float csum[NUM_CLASSES] = {0.f, 0.f, 0.f, 0.f, 0.f};
    float ccnt[NUM_CLASSES] = {0.f, 0.f, 0.f, 0.f, 0.f};

    const int tid     = blockIdx.x * BLOCK + threadIdx.x;
    const int stride  = GRID * BLOCK;
    const int nGroups = N >> 2;

    const float4* lp = (const float4*)logits;
    const int4*   tp = (const int4*)targets;

    for (int g = tid; g < nGroups; g += stride) {
        const float4* p = lp + (size_t)g * 5;
        // CDNA5 speculative prefetch of this lane's next group
        __builtin_prefetch(p + (size_t)stride * 5, 0, 1);

        float4 q0 = p[0], q1 = p[1], q2 = p[2], q3 = p[3], q4 = p[4];
        int4   t4 = tp[g];

        float r[4][NUM_CLASSES] = {
            {q0.x, q0.y, q0.z, q0.w, q1.x},
            {q1.y, q1.z, q1.w, q2.x, q2.y},
            {q2.z, q2.w, q3.x, q3.y, q3.z},
            {q3.w, q4.x, q4.y, q4.z, q4.w}};
        int tg[4] = {t4.x, t4.y, t4.z, t4.w};

#pragma unroll
        for (int j = 0; j < 4; ++j) {
            float a0 = r[j][0], a1 = r[j][1], a2 = r[j][2], a3 = r[j][3], a4 = r[j][4];
            int   t  = tg[j];

            float m  = fmaxf(fmaxf(fmaxf(a0, a1), fmaxf(a2, a3)), a4);
            float s  = __expf(a0 - m) + __expf(a1 - m) + __expf(a2 - m) +
                       __expf(a3 - m) + __expf(a4 - m);
            float xt = a0;
            xt = (t == 1) ? a1 : xt;
            xt = (t == 2) ? a2 : xt;
            xt = (t == 3) ? a3 : xt;
            xt = (t == 4) ? a4 : xt;
            float loss = __logf(s) + m - xt;   // -log_softmax[target]

#pragma unroll
            for (int c = 0; c < NUM_CLASSES; ++c) {
                bool hit = (t == c);
                csum[c] += hit ? loss : 0.f;
                ccnt[c] += hit ? 1.f  : 0.f;
            }
        }
    }

    // scalar tail (empty when N % 4 == 0)
    for (int i = (nGroups << 2) + tid; i < N; i += stride) {
        const float* row = logits + (size_t)i * NUM_CLASSES;
        float a0 = row[0], a1 = row[1], a2 = row[2], a3 = row[3], a4 = row[4];
        int   t  = targets[i];
        float m  = fmaxf(fmaxf(fmaxf(a0, a1), fmaxf(a2, a3)), a4);
        float s  = __expf(a0 - m) + __expf(a1 - m) + __expf(a2 - m) +
                   __expf(a3 - m) + __expf(a4 - m);
        float xt = a0;
        xt = (t == 1) ? a1 : xt;
        xt = (t == 2) ? a2 : xt;
        xt = (t == 3) ? a3 : xt;
        xt = (t == 4) ? a4 : xt;
        float loss = __logf(s) + m - xt;
#pragma unroll
        for (int c = 0; c < NUM_CLASSES; ++c) {
            bool hit = (t == c);
            csum[c] += hit ? loss : 0.f;
            ccnt[c] += hit ? 1.f  : 0.f;
        }
    }

    // wave32 butterfly reduction
#pragma unroll
    for (int c = 0; c < NUM_CLASSES; ++c) {
#pragma unroll
        for (int off = 16; off > 0; off >>= 1) {
            csum[c] += __shfl_xor(csum[c], off, 32);
            ccnt[c] += __shfl_xor(ccnt[c], off, 32);
        }
    }

    __shared__ float acc[16];
    if (threadIdx.x < 16) acc[threadIdx.x] = 0.f;
    __syncthreads();

    if ((threadIdx.x & 31) == 0) {  // one lane per wave -> ds_add_f32
#pragma unroll
        for (int c = 0; c < NUM_CLASSES; ++c) {
            atomicAdd(&acc[c],               csum[c]);
            atomicAdd(&acc[NUM_CLASSES + c], ccnt[c]);
        }
    }
    __syncthreads();

    if (threadIdx.x < 16)
        partials[blockIdx.x * 16 + threadIdx.x] = acc[threadIdx.x];
}

// Stage 2: one wave. Column-sum of partials[numRows][16] via
// V_WMMA_F32_16X16X4_F32 with A = ones(16x4):  D += ones * P_chunk(4x16).
// B layout (4x16 f32, 2 VGPRs): VGPR0 = {K0 | K2}, VGPR1 = {K1 | K3}, N = lane&15.
__global__ void __launch_bounds__(32) mfe_final_kernel(
    const float* __restrict__ partials, float* __restrict__ out, int numRows)
{
    const int lane  = threadIdx.x;
    const int n     = lane & 15;
    const int kbase = (lane < 16) ? 0 : 2;

    v2f a;
    a.x = 1.0f;
    a.y = 1.0f;
    v8f c = {};

    for (int r = 0; r < numRows; r += 4) {
        v2f b;
        b.x = partials[(r + kbase    ) * 16 + n];
        b.y = partials[(r + kbase + 1) * 16 + n];
        c = __builtin_amdgcn_wmma_f32_16x16x4_f32(
                /*neg_a=*/false, a, /*neg_b=*/false, b,
                /*c_mod=*/(short)0, c, /*reuse_a=*/false, /*reuse_b=*/false);
    }

    // All D rows are identical (A rows identical): every lane's c[0] holds the
    // column total for column (lane & 15).
    float col = c[0];
    float cnt = __shfl(col, n + NUM_CLASSES, 32);  // count column paired with sum column n

    float mean = 0.f;
    if (lane < NUM_CLASSES)
        mean = (cnt > 0.f) ? (col / cnt) : 0.f;

#pragma unroll
    for (int off = 16; off > 0; off >>= 1)
        mean += __shfl_xor(mean, off, 32);

    if (lane == 0)
        out[0] = mean;
}

extern "C" void kernel_launch(void* const* d_in, const int* in_sizes, int n_in,
                              void* d_out, int out_size, void* d_ws, size_t ws_size,
                              hipStream_t stream) {
    const float* logits  = (const float*)d_in[0];
    const int*   targets = (const int*)d_in[1];
    float*       out     = (float*)d_out;
    float*       parts   = (float*)d_ws;     // GRID * 16 floats = 64 KB
    const int    N       = in_sizes[1];      // targets count

    mfe_partials_kernel<<<GRID, BLOCK, 0, stream>>>(logits, targets, parts, N);
    mfe_final_kernel<<<1, 32, 0, stream>>>(parts, out, GRID);
}